// attention_decoding_56495999811906
// MI455X (gfx1250) — compile-verified
//
#include <hip/hip_runtime.h>
#include <stdint.h>
#include <math.h>

// Problem constants (from reference): B=128, F=256, N=2000, scale = 1/sqrt(F)
#define BB 128
#define FF 256
#define NN 2000
#define SCALE 0.0625f

#define TILE_N 128          // nodes per LDS tile in pass B (256*128*4 = 128 KB LDS)
#define PB_THREADS 1024     // 32 waves per workgroup in pass B

typedef __attribute__((ext_vector_type(2))) float v2f;
typedef __attribute__((ext_vector_type(8))) float v8f;

// ---------------------------------------------------------------------------
// CDNA5 async global->LDS copy (ASYNCcnt path, §15.18 opcode 97) via inline asm
// ---------------------------------------------------------------------------
__device__ __forceinline__ void async_copy_b64(const void* gp, void* lp) {
#if defined(__HIP_DEVICE_COMPILE__)
  unsigned int loff = (unsigned int)(uintptr_t)lp;      // generic LDS ptr: low 32 bits = LDS addr
  unsigned long long ga = (unsigned long long)(uintptr_t)gp;
  asm volatile("global_load_async_to_lds_b64 %0, %1, off"
               :: "v"(loff), "v"(ga) : "memory");
#else
  (void)gp; (void)lp;
#endif
}

__device__ __forceinline__ void wait_asynccnt0() {
#if defined(__HIP_DEVICE_COMPILE__)
  asm volatile("s_wait_asynccnt 0" ::: "memory");
#endif
}

// ---------------------------------------------------------------------------
// Pass A: glb[b,f] = max_n x[b,f,n]   (one wave per row, float4 loads)
// ---------------------------------------------------------------------------
__global__ __launch_bounds__(256) void rowmax_kernel(const float* __restrict__ x,
                                                     float* __restrict__ glb) {
  const int row  = blockIdx.x * 8 + (threadIdx.x >> 5);   // row in [0, B*F)
  const int lane = threadIdx.x & 31;
  const float4* rp = (const float4*)(x + (size_t)row * NN);
  float m = -INFINITY;
  for (int c = lane; c < NN / 4; c += 32) {               // 2000/4 = 500 exact
    float4 v = rp[c];
    m = fmaxf(m, fmaxf(fmaxf(v.x, v.y), fmaxf(v.z, v.w)));
  }
#pragma unroll
  for (int o = 16; o; o >>= 1) m = fmaxf(m, __shfl_xor(m, o));
  if (lane == 0) glb[row] = m;
}

// ---------------------------------------------------------------------------
// Gather last/current city features: context[b, 0:F]=x[:, last], [F:2F]=x[:, cur]
// ---------------------------------------------------------------------------
__global__ __launch_bounds__(256) void gather_kernel(const float* __restrict__ x,
                                                     const int* __restrict__ lc,
                                                     float* __restrict__ ctx) {
  const int id = blockIdx.x * 256 + threadIdx.x;          // B*2F total
  const int b = id / (2 * FF);
  const int k = id - b * 2 * FF;
  const int j = (k >= FF) ? 1 : 0;
  const int f = (k >= FF) ? k - FF : k;
  const int n = lc[b * 2 + j];
  ctx[id] = x[((size_t)b * FF + f) * NN + n];
}

// ---------------------------------------------------------------------------
// WMMA f32 GEMM: C[M,N] = A[M,K] @ B + bias + addm
//   B element (k,n) = Bp[k*sbk + n*sbn]  (handles W / W^T via strides)
// One wave per 16x16 output tile, V_WMMA_F32_16X16X4_F32, K stepped by 4.
// VGPR layouts per ISA §7.12.2 (wave32):
//   A 16x4 : lane<16 -> (M=lane, K=k+0..1), lane>=16 -> (M=lane-16, K=k+2..3)
//   B 4x16 : VGPR0 = rows k+0 / k+2, VGPR1 = rows k+1 / k+3
//   C 16x16: VGPR r -> rows r (lanes 0-15) and r+8 (lanes 16-31)
// ---------------------------------------------------------------------------
__global__ __launch_bounds__(32) void gemm16_kernel(
    const float* __restrict__ A, int lda,
    const float* __restrict__ Bp, int sbk, int sbn,
    const float* __restrict__ bias, const float* __restrict__ addm,
    float* __restrict__ C, int ldc, int ntiles_n, int K) {
  const int m0  = (blockIdx.x / ntiles_n) << 4;
  const int n0  = (blockIdx.x % ntiles_n) << 4;
  const int lane = threadIdx.x & 31;
  const int l16 = lane & 15;
  const int hi  = lane >> 4;
  v8f acc = {};
#if __has_builtin(__builtin_amdgcn_wmma_f32_16x16x4_f32)
  for (int k = 0; k < K; k += 4) {
    v2f a, bv;
    const float* ap = A + (size_t)(m0 + l16) * lda + (k + 2 * hi);
    a[0] = ap[0];
    a[1] = ap[1];
    const int kk = k + 2 * hi;
    const size_t nb = (size_t)(n0 + l16) * sbn;
    bv[0] = Bp[(size_t)kk * sbk + nb];
    bv[1] = Bp[(size_t)(kk + 1) * sbk + nb];
    acc = __builtin_amdgcn_wmma_f32_16x16x4_f32(
        false, a, false, bv, (short)0, acc, false, false);
  }
#else
  for (int r = 0; r < 8; ++r) {
    const int row = m0 + r + 8 * hi;
    float s = 0.f;
    for (int k = 0; k < K; ++k)
      s += A[(size_t)row * lda + k] * Bp[(size_t)k * sbk + (size_t)(n0 + l16) * sbn];
    acc[r] = s;
  }
#endif
  const int col = n0 + l16;
  const float bcol = bias ? bias[col] : 0.f;
#pragma unroll
  for (int r = 0; r < 8; ++r) {
    const int row = m0 + r + 8 * hi;
    float v = acc[r] + bcol;
    if (addm) v += addm[(size_t)row * ldc + col];
    C[(size_t)row * ldc + col] = v;
  }
}

// ---------------------------------------------------------------------------
// out[b] = sum_f A[b,f] * v[f]   (one wave per batch row)
// ---------------------------------------------------------------------------
__global__ __launch_bounds__(32) void dotrow_kernel(const float* __restrict__ A,
                                                    const float* __restrict__ v,
                                                    float* __restrict__ out, int K) {
  const int b = blockIdx.x;
  const int lane = threadIdx.x;
  float s = 0.f;
  for (int f = lane; f < K; f += 32) s += A[(size_t)b * K + f] * v[f];
#pragma unroll
  for (int o = 16; o; o >>= 1) s += __shfl_xor(s, o);
  if (lane == 0) out[b] = s;
}

// ---------------------------------------------------------------------------
// Pass B: per batch, fused scores + online softmax + xbar accumulation.
// x[b] is streamed through LDS tiles exactly once via async global->LDS copies.
//   t[n]   = (qk . x[:,n] + cK) * scale
//   attn   = softmax(t)          (online, flash-style rescaling)
//   xbar[f]= sum_n attn[n] * x[f,n]
// ---------------------------------------------------------------------------
__global__ __launch_bounds__(PB_THREADS) void passB_kernel(
    const float* __restrict__ x, const float* __restrict__ qk,
    const float* __restrict__ cK, float* __restrict__ xbar) {
  __shared__ float tile[FF * TILE_N];   // 128 KB x-tile
  __shared__ float qs[FF];
  __shared__ float abuf[TILE_N];
  __shared__ float xb[FF];
  __shared__ float p1[PB_THREADS];
  __shared__ float red[32];
  __shared__ float ms[2];               // running max, running sum
  const int b = blockIdx.x;
  const int tid = threadIdx.x;
  const float* xbase = x + (size_t)b * FF * NN;
  if (tid < FF) { qs[tid] = qk[b * FF + tid]; xb[tid] = 0.f; }
  if (tid == 0) { ms[0] = -INFINITY; ms[1] = 0.f; }
  const float cKb = cK[b];
  __syncthreads();

  for (int t0 = 0; t0 < NN; t0 += TILE_N) {
    const int nt = min(TILE_N, NN - t0);        // 128 or final 80 (both even)
    // --- stream tile into LDS: async b64 copies, tracked by ASYNCcnt ---
    const int chunksPerRow = nt >> 1;           // b64 = 2 floats
    const int chunks = FF * chunksPerRow;
    for (int c = tid; c < chunks; c += PB_THREADS) {
      const int f = c / chunksPerRow;
      const int i = c - f * chunksPerRow;
      async_copy_b64(xbase + (size_t)f * NN + t0 + 2 * i,
                     &tile[f * TILE_N + 2 * i]);
    }
    wait_asynccnt0();
    __syncthreads();

    // --- scores: split F across 8 thread groups, then combine ---
    {
      const int n = tid & (TILE_N - 1);
      const int g = tid >> 7;                   // 0..7
      const int f0 = g * (FF / 8);
      float acc = 0.f;
      for (int f = f0; f < f0 + FF / 8; ++f)
        acc += qs[f] * tile[f * TILE_N + n];
      p1[tid] = acc;
    }
    __syncthreads();
    float tval = -INFINITY;
    if (tid < TILE_N && tid < nt) {
      float s = 0.f;
#pragma unroll
      for (int g = 0; g < 8; ++g) s += p1[g * TILE_N + tid];
      tval = (s + cKb) * SCALE;
    }

    // --- block max of tval ---
    float wm = tval;
#pragma unroll
    for (int o = 16; o; o >>= 1) wm = fmaxf(wm, __shfl_xor(wm, o));
    if ((tid & 31) == 0) red[tid >> 5] = wm;
    __syncthreads();
    if (tid < 32) {
      float v = red[tid];
#pragma unroll
      for (int o = 16; o; o >>= 1) v = fmaxf(v, __shfl_xor(v, o));
      if (tid == 0) red[0] = v;
    }
    __syncthreads();
    const float tmax  = red[0];
    const float m_old = ms[0];
    const float m_new = fmaxf(m_old, tmax);
    const float r     = __expf(m_old - m_new);  // 0 on first tile (m_old=-inf)

    float e = 0.f;
    if (tid < TILE_N) {
      e = (tid < nt) ? __expf(tval - m_new) : 0.f;
      abuf[tid] = e;
    }
    __syncthreads();                            // abuf ready; red[0]/ms[0] consumed

    // --- block sum of e; thread 0 updates running (m, s) ---
    float wsum = e;
#pragma unroll
    for (int o = 16; o; o >>= 1) wsum += __shfl_xor(wsum, o);
    if ((tid & 31) == 0) red[tid >> 5] = wsum;
    __syncthreads();
    if (tid == 0) {
      float tsum = 0.f;
      for (int w = 0; w < 32; ++w) tsum += red[w];
      ms[1] = ms[1] * r + tsum;
      ms[0] = m_new;
    }

    // --- xbar accumulate: 4 threads per feature row, quad shuffle-combine ---
    {
      const int f = tid >> 2;
      const int j = tid & 3;
      float part = 0.f;
      for (int i = j; i < nt; i += 4)
        part += abuf[i] * tile[f * TILE_N + i];
      part += __shfl_xor(part, 1);
      part += __shfl_xor(part, 2);
      if (j == 0) xb[f] = xb[f] * r + part;
    }
    __syncthreads();
  }
  const float s = ms[1];
  if (tid < FF) xbar[b * FF + tid] = xb[tid] / s;   // softmax-normalized
}

// ---------------------------------------------------------------------------
// Pass C: logits[b,n] = tanh((ql . x[:,n] + cL) * scale) * 10
// 256 threads cover 256 consecutive n -> coalesced row walks; sw prefetch ahead.
// ---------------------------------------------------------------------------
__global__ __launch_bounds__(256) void passC_kernel(
    const float* __restrict__ x, const float* __restrict__ ql,
    const float* __restrict__ cL, float* __restrict__ out) {
  __shared__ float qs[FF];
  const int b = blockIdx.x >> 3;
  const int chunk = blockIdx.x & 7;
  const int tid = threadIdx.x;
  qs[tid] = ql[b * FF + tid];
  __syncthreads();
  const int n = chunk * 256 + tid;
  const float* xbase = x + (size_t)b * FF * NN;
  if (n < NN) {
    float acc = 0.f;
    for (int f = 0; f < FF; ++f) {
      if (f + 8 < FF) __builtin_prefetch(xbase + (size_t)(f + 8) * NN + n, 0, 0);
      acc += qs[f] * xbase[(size_t)f * NN + n];
    }
    out[b * NN + n] = tanhf((acc + cL[b]) * SCALE) * 10.f;
  }
}

// ---------------------------------------------------------------------------
extern "C" void kernel_launch(void* const* d_in, const int* in_sizes, int n_in,
                              void* d_out, int out_size, void* d_ws, size_t ws_size,
                              hipStream_t stream) {
  const float* x     = (const float*)d_in[0];   // [B,F,N,1]
  const int*   lc    = (const int*)  d_in[1];   // [B,2]
  const float* Wg    = (const float*)d_in[2];   // [F,F]
  const float* bg    = (const float*)d_in[3];   // [F]
  const float* Wnode = (const float*)d_in[4];   // [3F,F]
  const float* bnode = (const float*)d_in[5];   // [3F]
  const float* Wlc   = (const float*)d_in[6];   // [F,2F]
  const float* blc   = (const float*)d_in[7];   // [F]
  const float* Wproj = (const float*)d_in[8];   // [F,F]
  const float* bproj = (const float*)d_in[9];   // [F]
  float* out = (float*)d_out;                   // [B,N]

  // workspace carve-up (~1.32 MB of f32)
  float* ws   = (float*)d_ws;
  float* glb  = ws;                         // [B,F]
  float* ctx  = glb  + BB * FF;             // [B,2F]
  float* Qb   = ctx  + BB * 2 * FF;         // [B,F]
  float* cQ   = Qb   + BB * FF;             // [B,F]
  float* qk   = cQ   + BB * FF;             // [B,F]
  float* cK   = qk   + BB * FF;             // [B]
  float* xbar = cK   + BB;                  // [B,F]
  float* v1   = xbar + BB * FF;             // [B,F]
  float* nc   = v1   + BB * FF;             // [B,F]
  float* ql   = nc   + BB * FF;             // [B,F]
  float* cL   = ql   + BB * FF;             // [B]

  const int gtiles = (BB / 16) * (FF / 16); // 128 WMMA tiles per small GEMM

  // Pass A: glb = max_n x
  rowmax_kernel<<<BB * FF / 8, 256, 0, stream>>>(x, glb);
  // context gather
  gather_kernel<<<(BB * 2 * FF) / 256, 256, 0, stream>>>(x, lc, ctx);
  // Q = glb @ Wg^T + bg            (B[k,n] = Wg[n,k]: sbk=1, sbn=F)
  gemm16_kernel<<<gtiles, 32, 0, stream>>>(glb, FF, Wg, 1, FF, bg, nullptr, Qb, FF, FF / 16, FF);
  // cQ = ctx @ Wlc^T + blc + Q     (B[k,n] = Wlc[n,k]: sbk=1, sbn=2F)
  gemm16_kernel<<<gtiles, 32, 0, stream>>>(ctx, 2 * FF, Wlc, 1, 2 * FF, blc, Qb, cQ, FF, FF / 16, 2 * FF);
  // qk = cQ @ Wnode_K              (B[k,n] = Wnode[k,n]: sbk=F, sbn=1)
  gemm16_kernel<<<gtiles, 32, 0, stream>>>(cQ, FF, Wnode, FF, 1, nullptr, nullptr, qk, FF, FF / 16, FF);
  // cK = cQ . bnode_K
  dotrow_kernel<<<BB, 32, 0, stream>>>(cQ, bnode, cK, FF);
  // Pass B: fused attention (single read of x)
  passB_kernel<<<BB, PB_THREADS, 0, stream>>>(x, qk, cK, xbar);
  // v1 = xbar @ Wnode_V^T + bnode_V  (B[k,n] = Wnode[F+n,k]: base F*F, sbk=1, sbn=F)
  gemm16_kernel<<<gtiles, 32, 0, stream>>>(xbar, FF, Wnode + (size_t)FF * FF, 1, FF, bnode + FF, nullptr, v1, FF, FF / 16, FF);
  // nc = v1 @ Wproj^T + bproj
  gemm16_kernel<<<gtiles, 32, 0, stream>>>(v1, FF, Wproj, 1, FF, bproj, nullptr, nc, FF, FF / 16, FF);
  // ql = nc @ Wnode_L              (base 2F*F, sbk=F, sbn=1)
  gemm16_kernel<<<gtiles, 32, 0, stream>>>(nc, FF, Wnode + (size_t)2 * FF * FF, FF, 1, nullptr, nullptr, ql, FF, FF / 16, FF);
  // cL = nc . bnode_L
  dotrow_kernel<<<BB, 32, 0, stream>>>(nc, bnode + 2 * FF, cL, FF);
  // Pass C: logits
  passC_kernel<<<BB * 8, 256, 0, stream>>>(x, ql, cL, out);
}